// vgg11_adder_87608742904030
// MI455X (gfx1250) — compile-verified
//
#include <hip/hip_runtime.h>
#include <stdint.h>

typedef float v2f __attribute__((ext_vector_type(2)));
typedef float v8f __attribute__((ext_vector_type(8)));
typedef unsigned int u32x4 __attribute__((ext_vector_type(4)));
typedef int i32x8 __attribute__((ext_vector_type(8)));
typedef int i32x4 __attribute__((ext_vector_type(4)));

#define BN_EPS 1e-5f

// ---------------------------------------------------------------------------
// TDM helper: 2D tile load (nrows rows of row_len f32, global row stride
// gstride f32) from global into LDS at byte offset lds_off. D# built per
// CDNA5 ISA 8.3/8.4 (count=1, type=2, data_size=4B).
// ---------------------------------------------------------------------------
__device__ __forceinline__ unsigned lds_byte_off(const void* p) {
  // LDS aperture: low 32 bits of the generic pointer are the LDS byte offset.
  return (unsigned)(uintptr_t)p;
}

__device__ __forceinline__ void tdm_load_tile_f32(const float* gaddr, unsigned lds_off,
                                                  int row_len, int nrows, int gstride) {
  unsigned long long ga = (unsigned long long)(uintptr_t)gaddr;
  u32x4 g0;
  g0[0] = 1u;                                    // count=1, is_restore=0, gather off
  g0[1] = lds_off;                               // lds_addr
  g0[2] = (unsigned)(ga & 0xFFFFFFFFu);          // global_addr[31:0]
  g0[3] = (unsigned)((ga >> 32) & 0x01FFFFFFu)   // global_addr[56:32]
          | (2u << 30);                          // type = 2 (image)
  unsigned td0 = (unsigned)gstride;              // tensor_dim0 (elements)
  unsigned td1 = (unsigned)nrows;                // tensor_dim1
  i32x8 g1;
  g1[0] = (int)(2u << 16);                       // workgroup_mask=0, data_size=2 (4B)
  g1[1] = (int)((td0 & 0xFFFFu) << 16);          // tensor_dim0[15:0] in [31:16]
  g1[2] = (int)(((td0 >> 16) & 0xFFFFu) | ((td1 & 0xFFFFu) << 16)); // dim0 hi | dim1 lo
  g1[3] = (int)(((td1 >> 16) & 0xFFFFu) | (((unsigned)row_len & 0xFFFFu) << 16)); // dim1 hi | tile_dim0
  g1[4] = (int)((unsigned)nrows & 0xFFFFu);      // tile_dim1 (tile_dim2 = 0)
  g1[5] = (int)td0;                              // tensor_dim0_stride[31:0]
  g1[6] = 0;                                     // stride[47:32] | dim1_stride[15:0]
  g1[7] = 0;
  i32x4 z4 = {0, 0, 0, 0};                       // groups 2/3 unused (<=2D)
  i32x8 z8 = {0, 0, 0, 0, 0, 0, 0, 0};
  __builtin_amdgcn_tensor_load_to_lds(g0, g1, z4, z4, z8, 0);
}

// ---------------------------------------------------------------------------
// Adder 3x3 layer, pad=1: out[b,o,p] = -sum_{c,3x3} |in - w|
// Block: 256 threads = 256 output pixels, 16 output channels per block.
// Weights staged through the TDM into double-buffered LDS.
// ---------------------------------------------------------------------------
__global__ void adder3x3_kernel(const float* __restrict__ in, const float* __restrict__ w,
                                float* __restrict__ out, int C, int O, int H, int W) {
  const int tid = threadIdx.x;
  const int HW = H * W;
  const int p = blockIdx.x * 256 + tid;
  const int b = blockIdx.z;
  const int obase = blockIdx.y * 16;
  const int y = p / W;
  const int x = p - y * W;
  const bool interior = (p < HW) && (y > 0) && (y < H - 1) && (x > 0) && (x < W - 1);

  __shared__ float wsm[2][8 * 16 * 9];           // 2 x 1152 f32 = 9216 B

  float acc[16];
#pragma unroll
  for (int i = 0; i < 16; i++) acc[i] = 0.f;

  const int grow = C * 9;                        // global row stride per o (elements)

  // prefetch chunk 0 -> buffer 0 (one wave issues; TDM ignores EXEC)
  if (tid < 32) {
    int cc0 = (C < 8) ? C : 8;
    tdm_load_tile_f32(w + (size_t)obase * grow, lds_byte_off(&wsm[0][0]), cc0 * 9, 16, grow);
  }

  int buf = 0;
  for (int c0 = 0; c0 < C; c0 += 8, buf ^= 1) {
    const int cc = (C - c0 < 8) ? (C - c0) : 8;
    if (tid < 32) __builtin_amdgcn_s_wait_tensorcnt(0);
    __syncthreads();                             // staged weights visible to all waves
    const int c1 = c0 + 8;
    if (c1 < C && tid < 32) {
      int ccn = (C - c1 < 8) ? (C - c1) : 8;
      tdm_load_tile_f32(w + (size_t)obase * grow + (size_t)c1 * 9,
                        lds_byte_off(&wsm[buf ^ 1][0]), ccn * 9, 16, grow);
    }
    const float* ws = &wsm[buf][0];

    auto process = [&](int ccv) {                // called with literal -> const-prop rl
      const int rl = ccv * 9;                    // LDS row length per o
      for (int ci = 0; ci < ccv; ci++) {
        const int c = c0 + ci;
        const float* ip = in + ((size_t)b * C + c) * HW;
        if (c + 1 < C && p < HW)
          __builtin_prefetch(ip + HW + y * W, 0, 1);   // next channel row
        float v[9];
        if (interior) {
          const float* rp = ip + (size_t)(y - 1) * W + (x - 1);
#pragma unroll
          for (int dy = 0; dy < 3; dy++)
#pragma unroll
            for (int dx = 0; dx < 3; dx++)
              v[dy * 3 + dx] = rp[dy * W + dx];
        } else {
#pragma unroll
          for (int dy = 0; dy < 3; dy++)
#pragma unroll
            for (int dx = 0; dx < 3; dx++) {
              int yy = y + dy - 1, xx = x + dx - 1;
              bool ok = (p < HW) && (yy >= 0) && (yy < H) && (xx >= 0) && (xx < W);
              v[dy * 3 + dx] = ok ? ip[yy * W + xx] : 0.f;  // zero-pad: |0-w| = |w|
            }
        }
#pragma unroll
        for (int o = 0; o < 16; o++) {
          const float* wp = ws + o * rl + ci * 9;
          float s = 0.f;
#pragma unroll
          for (int j = 0; j < 9; j++) s += fabsf(v[j] - wp[j]);
          acc[o] -= s;
        }
      }
    };
    if (cc == 8) process(8); else process(cc);
  }
  if (p < HW) {
#pragma unroll
    for (int o = 0; o < 16; o++)
      out[((size_t)b * O + obase + o) * HW + p] = acc[o];
  }
}

// ---------------------------------------------------------------------------
// BN statistics via V_WMMA_F32_16X16X4_F32: D = A(16ch x 4elem) x ones(4x16)+C
// Every column of D holds the per-channel running sum. Two WMMA chains: sum
// and sum-of-squares. 16 channels per wave; slices are per-image and 4-aligned
// (HW is always a multiple of 4) so the loop has no guards and no division:
// one global_load_b64 + v_dual_mul + 2 x WMMA per 4 elements x 16 channels.
// Grid: (C/16, 8); blockIdx.y -> {image b = y>>2, slice group y&3}; 8 waves.
// ---------------------------------------------------------------------------
__global__ void bn_stats_wmma_kernel(const float* __restrict__ yv, float* __restrict__ stats,
                                     int C, int HW) {
  const int lane = threadIdx.x & 31;
  const int wave = threadIdx.x >> 5;
  const int cbase = blockIdx.x * 16;
  const int m = lane & 15;                       // channel row of A
  const int half = lane >> 4;                    // K group (0 -> K0/1, 1 -> K2/3)
  const int bb = blockIdx.y >> 2;                // image
  const int sl = (blockIdx.y & 3) * 8 + wave;    // 32 slices per image

  const float* base = yv + ((size_t)bb * C + (cbase + m)) * HW;

  const int per = ((HW + 127) >> 7) << 2;        // ceil(HW/128)*4, multiple of 4
  int s0 = sl * per;
  int s1 = s0 + per; if (s1 > HW) s1 = HW;       // 4-aligned (HW % 4 == 0)

  v8f csum = {}; v8f csq = {};
  v2f ones = {1.f, 1.f};

  const v2f* bp = (const v2f*)(base) + (s0 >> 1) + half;
  for (int e0 = s0; e0 < s1; e0 += 4, bp += 2) {
    v2f a  = *bp;                                // elements e0+2*half, e0+2*half+1
    v2f a2 = a * a;
    csum = __builtin_amdgcn_wmma_f32_16x16x4_f32(false, a,  false, ones, (short)0, csum, false, false);
    csq  = __builtin_amdgcn_wmma_f32_16x16x4_f32(false, a2, false, ones, (short)0, csq,  false, false);
  }
  // D layout: VGPR j -> row M=j (lanes 0-15) / M=j+8 (lanes 16-31); cols equal.
  if (lane == 0 || lane == 16) {
    int mb = (lane == 0) ? 0 : 8;
#pragma unroll
    for (int j = 0; j < 8; j++) {
      int ch = cbase + mb + j;
      atomicAdd(&stats[ch], csum[j]);
      atomicAdd(&stats[C + ch], csq[j]);
    }
  }
}

// Grid: ((HW+255)/256, C, B) -- no per-thread div/mod.
__global__ void bn_apply_kernel(float* __restrict__ yv, const float* __restrict__ stats,
                                const float* __restrict__ g, const float* __restrict__ b,
                                int C, int HW, float invN) {
  int p = blockIdx.x * 256 + threadIdx.x;
  if (p >= HW) return;
  int c = blockIdx.y;
  float mu = stats[c] * invN;
  float var = stats[C + c] * invN - mu * mu;
  size_t i = ((size_t)blockIdx.z * C + c) * HW + p;
  yv[i] = (yv[i] - mu) * rsqrtf(var + BN_EPS) * g[c] + b[c];
}

// Grid: ((HoWo+255)/256, C, B)
__global__ void pool_relu_kernel(const float* __restrict__ in, float* __restrict__ out,
                                 int C, int H, int W) {
  int Wo = W >> 1, Ho = H >> 1;
  int p = blockIdx.x * 256 + threadIdx.x;
  if (p >= Ho * Wo) return;
  int c = blockIdx.y, b = blockIdx.z;
  int y = p / Wo, x = p - y * Wo;
  const float* src = in + (((size_t)b * C + c) * H + y * 2) * W + x * 2;
  float m = fmaxf(fmaxf(src[0], src[1]), fmaxf(src[W], src[W + 1]));
  out[((size_t)b * C + c) * (Ho * Wo) + p] = fmaxf(m, 0.f);
}

__global__ void relu_kernel(const float* __restrict__ in, float* __restrict__ out, int total) {
  int i = blockIdx.x * blockDim.x + threadIdx.x;
  if (i >= total) return;
  out[i] = fmaxf(in[i], 0.f);
}

__global__ void global_max_kernel(const float* __restrict__ in, float* __restrict__ out,
                                  int HW, int total) {
  int i = blockIdx.x * blockDim.x + threadIdx.x;
  if (i >= total) return;
  const float* p = in + (size_t)i * HW;
  float m = -3.402823466e38f;
  for (int j = 0; j < HW; j++) m = fmaxf(m, p[j]);
  out[i] = m;
}

// Final 1x1 adder "linear" (512 -> 1000) + training-mode BN over the batch of 2.
__global__ void fc_adder_bn_kernel(const float* __restrict__ vec, const float* __restrict__ wl,
                                   const float* __restrict__ gl, const float* __restrict__ bl,
                                   float* __restrict__ out) {
  const int o = blockIdx.x;
  const int tid = threadIdx.x;
  __shared__ float red0[256];
  __shared__ float red1[256];
  float s0 = 0.f, s1 = 0.f;
  for (int c = tid; c < 512; c += 256) {
    float wv = wl[o * 512 + c];
    s0 -= fabsf(vec[c] - wv);
    s1 -= fabsf(vec[512 + c] - wv);
  }
  red0[tid] = s0; red1[tid] = s1;
  __syncthreads();
  for (int st = 128; st > 0; st >>= 1) {
    if (tid < st) { red0[tid] += red0[tid + st]; red1[tid] += red1[tid + st]; }
    __syncthreads();
  }
  if (tid == 0) {
    float p0 = red0[0], p1 = red1[0];
    float mu = 0.5f * (p0 + p1);
    float var = 0.5f * ((p0 - mu) * (p0 - mu) + (p1 - mu) * (p1 - mu));
    float is = rsqrtf(var + BN_EPS) * gl[o];
    out[o]        = (p0 - mu) * is + bl[o];
    out[1000 + o] = (p1 - mu) * is + bl[o];
  }
}

// ---------------------------------------------------------------------------
extern "C" void kernel_launch(void* const* d_in, const int* in_sizes, int n_in,
                              void* d_out, int out_size, void* d_ws, size_t ws_size,
                              hipStream_t stream) {
  (void)in_sizes; (void)n_in; (void)out_size; (void)ws_size;
  const float* x = (const float*)d_in[0];
  const float* Wt[8]; const float* G[8]; const float* Bb[8];
  for (int i = 0; i < 8; i++) {
    Wt[i] = (const float*)d_in[1 + 3 * i];
    G[i]  = (const float*)d_in[2 + 3 * i];
    Bb[i] = (const float*)d_in[3 + 3 * i];
  }
  const float* wl = (const float*)d_in[25];
  const float* gl = (const float*)d_in[26];
  const float* bl = (const float*)d_in[27];
  float* out = (float*)d_out;

  // Workspace layout (floats): actA | actB | stats(1024) | vec(1024)
  float* actA  = (float*)d_ws;
  float* actB  = actA + 294912;
  float* stats = actB + 294912;
  float* vec   = stats + 1024;

  static const int  Cs[8]   = {3, 64, 128, 256, 256, 512, 512, 512};
  static const int  Os[8]   = {64, 128, 256, 256, 512, 512, 512, 512};
  static const bool pool[8] = {true, true, false, true, false, true, false, true};

  int H = 96, Wd = 96;
  const float* cur = x;
  float* feat = out + 2000;                      // feats follow the 2x1000 logits
  float* nxt = actA; float* other = actB;

  for (int i = 0; i < 8; i++) {
    const int C = Cs[i], O = Os[i], HW = H * Wd;
    adder3x3_kernel<<<dim3((HW + 255) / 256, O / 16, 2), 256, 0, stream>>>(
        cur, Wt[i], feat, C, O, H, Wd);
    (void)hipMemsetAsync(stats, 0, 2 * O * sizeof(float), stream);
    bn_stats_wmma_kernel<<<dim3(O / 16, 8), 256, 0, stream>>>(feat, stats, O, HW);
    const int total = 2 * O * HW;
    bn_apply_kernel<<<dim3((HW + 255) / 256, O, 2), 256, 0, stream>>>(
        feat, stats, G[i], Bb[i], O, HW, 1.0f / (float)(2 * HW));
    if (pool[i]) {
      const int Ho = H / 2, Wo = Wd / 2;
      pool_relu_kernel<<<dim3((Ho * Wo + 255) / 256, O, 2), 256, 0, stream>>>(feat, nxt, O, H, Wd);
      H = Ho; Wd = Wo;
    } else {
      relu_kernel<<<(total + 255) / 256, 256, 0, stream>>>(feat, nxt, total);
    }
    cur = nxt;
    float* t = nxt; nxt = other; other = t;
    feat += total;
  }

  // cur: [2,512,3,3] -> adaptive max -> [2,512]
  global_max_kernel<<<(1024 + 255) / 256, 256, 0, stream>>>(cur, vec, H * Wd, 1024);
  fc_adder_bn_kernel<<<1000, 256, 0, stream>>>(vec, wl, gl, bl, out);
}